// ScoreLabelSmoothedCrossEntropyLoss_27187142983911
// MI455X (gfx1250) — compile-verified
//
#include <hip/hip_runtime.h>
#include <hip/hip_bf16.h>

#ifndef __has_builtin
#define __has_builtin(x) 0
#endif

#if __has_builtin(__builtin_amdgcn_global_load_async_to_lds_b128)
#define HAVE_ASYNC_LDS 1
#else
#define HAVE_ASYNC_LDS 0
#endif

typedef float v4f __attribute__((ext_vector_type(4)));
typedef int   v4i __attribute__((ext_vector_type(4)));
typedef __attribute__((address_space(1))) v4i* as1_v4i;   // global (device) int4*
typedef __attribute__((address_space(3))) v4i* as3_v4i;   // LDS int4*

#define NCLS   32000
#define NV4    (NCLS / 4)                 // 8000 float4 per row
#define TPB    256
#define NSTAGE ((NV4 + TPB - 1) / TPB)    // 32 stages (last partial: 64 lanes)
#define PIPE   4                          // async pipeline depth (quad buffer)
#define CPOL_NT 1                         // load temporal hint: NT (stream-once)

__device__ __forceinline__ float exp2_fast(float x) {
#if __has_builtin(__builtin_amdgcn_exp2f)
  return __builtin_amdgcn_exp2f(x);       // v_exp_f32
#else
  return exp2f(x);
#endif
}
__device__ __forceinline__ float log2_fast(float x) {
#if __has_builtin(__builtin_amdgcn_logf)
  return __builtin_amdgcn_logf(x);        // v_log_f32 (log2)
#else
  return log2f(x);
#endif
}

#if __has_builtin(__builtin_amdgcn_s_wait_asynccnt)
#define WAIT_ASYNC(N) do { __builtin_amdgcn_s_wait_asynccnt(N); \
                           asm volatile("" ::: "memory"); } while (0)
#else
#define WAIT_ASYNC(N) asm volatile("s_wait_asynccnt " #N ::: "memory")
#endif

#if HAVE_ASYNC_LDS
__device__ __forceinline__ void async_load_b128(const v4f* g, v4f* l) {
  __builtin_amdgcn_global_load_async_to_lds_b128(
      (as1_v4i)(v4i*)(v4f*)g,     // drop const, reinterpret, addrspacecast -> AS1
      (as3_v4i)(v4i*)l,           // reinterpret, addrspacecast -> AS3
      0, CPOL_NT);
}
#endif

__global__ __launch_bounds__(TPB)
void row_loss_kernel(const float* __restrict__ yhat,
                     const int*   __restrict__ y,
                     float*       __restrict__ row_loss)
{
  const int row   = blockIdx.x;
  const int tid   = threadIdx.x;
  const int t     = y[row];        // uniform per block -> scalar load
  const int par   = t & 1;
  const int tIdx4 = t >> 2;        // float4 slot holding the label logit
  const int tComp = t & 3;

  const v4f* rowp = (const v4f*)(yhat + (size_t)row * NCLS);

  __shared__ float sh_m[TPB];
  __shared__ float sh_s[TPB];
  __shared__ float sh_sel[TPB];
  __shared__ float sh_yt;

#if HAVE_ASYNC_LDS
  __shared__ v4f stage[PIPE][TPB];   // 16 KB quad buffer, 4 KB per stage
#endif

  const float LOG2E = 1.4426950408889634f;
  float m   = -__builtin_inff();   // online-LSE running max
  float s   = 0.0f;                // online-LSE running sum (scaled by exp(-m))
  float sel = 0.0f;                // parity-selected plain sum
  float yt  = 0.0f;
  bool  has_yt = false;

#if HAVE_ASYNC_LDS
  // Prologue: put PIPE-1 stages in flight. Addresses are clamped so every
  // lane of every wave always issues -> ASYNCcnt is uniform per wave and the
  // wait immediates below are exact.
  for (int st = 0; st < PIPE - 1; ++st) {
    int gi = st * TPB + tid; if (gi > NV4 - 1) gi = NV4 - 1;
    async_load_b128(rowp + gi, &stage[st][tid]);
  }
#endif

  for (int st = 0; st < NSTAGE; ++st) {
    const int sidx = st * TPB + tid;
#if HAVE_ASYNC_LDS
    // Always issue stage st+PIPE-1 (clamped; tail issues are redundant loads
    // into buffers never read again). Keeps exactly PIPE loads outstanding.
    {
      int gi = (st + PIPE - 1) * TPB + tid; if (gi > NV4 - 1) gi = NV4 - 1;
      async_load_b128(rowp + gi, &stage[(st + PIPE - 1) & (PIPE - 1)][tid]);
    }
    WAIT_ASYNC(PIPE - 1);   // in-order completion: oldest (stage st) is done
    v4f v = stage[st & (PIPE - 1)][tid];
#else
    v4f v = {0.f, 0.f, 0.f, 0.f};
    if (sidx < NV4) v = __builtin_nontemporal_load(rowp + sidx);
#endif
    if (sidx < NV4) {
      // element base = 4*sidx (even): components 0,2 are even classes; 1,3 odd
      sel += (par == 0) ? (v[0] + v[2]) : (v[1] + v[3]);
      if (sidx == tIdx4) { yt = v[tComp]; has_yt = true; }
      float mx = fmaxf(fmaxf(v[0], v[1]), fmaxf(v[2], v[3]));
      float nm = fmaxf(m, mx);
      float r  = exp2_fast((m    - nm) * LOG2E);
      float e0 = exp2_fast((v[0] - nm) * LOG2E);
      float e1 = exp2_fast((v[1] - nm) * LOG2E);
      float e2 = exp2_fast((v[2] - nm) * LOG2E);
      float e3 = exp2_fast((v[3] - nm) * LOG2E);
      s = s * r + ((e0 + e1) + (e2 + e3));
      m = nm;
    }
  }

  if (has_yt) sh_yt = yt;          // exactly one thread owns the label slot
  sh_m[tid] = m; sh_s[tid] = s; sh_sel[tid] = sel;
  __syncthreads();

  for (int off = TPB >> 1; off > 0; off >>= 1) {
    if (tid < off) {
      float m1 = sh_m[tid], s1 = sh_s[tid];
      float m2 = sh_m[tid + off], s2 = sh_s[tid + off];
      float nm = fmaxf(m1, m2);
      sh_s[tid] = s1 * exp2_fast((m1 - nm) * LOG2E)
                + s2 * exp2_fast((m2 - nm) * LOG2E);
      sh_m[tid] = nm;
      sh_sel[tid] += sh_sel[tid + off];
    }
    __syncthreads();
  }

  if (tid == 0) {
    const float smooth = 0.1f / (float)(NCLS / 2 - 1);
    float lse = sh_m[0] + 0.6931471805599453f * log2_fast(sh_s[0]);
    // loss_row = lse - smooth*SEL - (0.9 - smooth)*y_hat[t]
    row_loss[row] = lse - smooth * sh_sel[0] - (0.9f - smooth) * sh_yt;
  }
}

__global__ __launch_bounds__(TPB)
void mean_kernel(const float* __restrict__ in, float* __restrict__ out, int n)
{
  __shared__ float sh[TPB];
  float acc = 0.0f;
  for (int i = threadIdx.x; i < n; i += TPB) acc += in[i];
  sh[threadIdx.x] = acc;
  __syncthreads();
  for (int off = TPB >> 1; off > 0; off >>= 1) {
    if (threadIdx.x < off) sh[threadIdx.x] += sh[threadIdx.x + off];
    __syncthreads();
  }
  if (threadIdx.x == 0) out[0] = sh[0] * (1.0f / (float)n);
}

extern "C" void kernel_launch(void* const* d_in, const int* in_sizes, int n_in,
                              void* d_out, int out_size, void* d_ws, size_t ws_size,
                              hipStream_t stream)
{
  const float* yhat = (const float*)d_in[0];
  const int*   y    = (const int*)d_in[1];
  const int rows    = in_sizes[1];       // 4096
  float* row_loss   = (float*)d_ws;      // 4096 floats of scratch

  row_loss_kernel<<<rows, TPB, 0, stream>>>(yhat, y, row_loss);
  mean_kernel<<<1, TPB, 0, stream>>>(row_loss, (float*)d_out, rows);
}